// GroupGemmOp_67903432950460
// MI455X (gfx1250) — compile-verified
//
#include <hip/hip_runtime.h>
#include <hip/hip_bf16.h>

// Grouped fp32 GEMM for MI455X (gfx1250, wave32) using V_WMMA_F32_16X16X4_F32.
// 8 problems: C_g[M_g x 4096] = A_g[M_g x 4096] * B_g[4096 x 4096], fp32.
// Compute-bound in fp32 (AI ~508 FLOP/B vs 23.3 TB/s HBM).
//
// LDS holds A and B tiles in k-pair-interleaved float2 form so every WMMA
// fragment (which needs K and K+1 in an adjacent VGPR pair) is produced by a
// single conflict-free ds_load_b64. Fragments are double-buffered in registers
// so each group's LDS loads are issued a full WMMA-group ahead of their use.

typedef __attribute__((ext_vector_type(2))) float v2f;
typedef __attribute__((ext_vector_type(4))) float f4;
typedef __attribute__((ext_vector_type(8))) float v8f;

#define GG_K 4096
#define GG_N 4096
#define BM 128
#define BN 128
#define BK 32
#define P2 144                 // float2 pitch per k-pair row; 2*P2 % 64 == 32
#define PITCH (2 * P2)         // dword pitch (288): hi-half lanes hit disjoint banks
#define NKB (GG_K / BK)        // 128 k-blocks

struct GGArgs {
    const float* A[8];
    const float* B[8];
    unsigned long long outOff[8]; // element offset of each problem's C in d_out
    int tileBase[9];              // prefix sums of per-problem tile counts
};

__device__ __forceinline__ void load_tiles(const float* __restrict__ A,
                                           const float* __restrict__ B,
                                           int m0, int n0, int kt,
                                           int am, int akb, int bnq, int brp0,
                                           f4 ra[4], f4 rbl[2], f4 rbh[2]) {
#pragma unroll
    for (int i = 0; i < 4; ++i) {
        // A: row-panel column slice, 16B aligned (kt multiple of 32)
        size_t aidx = (size_t)(m0 + am) * GG_K + (size_t)(kt + (akb + 2 * i) * 4);
        ra[i] = *(const f4*)(A + aidx);
    }
#pragma unroll
    for (int i = 0; i < 2; ++i) {
        // B: two adjacent k-rows, each 512B/wave coalesced
        const int rp = brp0 + 8 * i;
        size_t bidx = (size_t)(kt + 2 * rp) * GG_N + (size_t)(n0 + bnq * 4);
        rbl[i] = *(const f4*)(B + bidx);
        rbh[i] = *(const f4*)(B + bidx + GG_N);
    }
}

__device__ __forceinline__ void store_tiles(float* __restrict__ Al, float* __restrict__ Bl,
                                            int am, int akb, int bnq, int brp0,
                                            const f4 ra[4], const f4 rbl[2], const f4 rbh[2]) {
#pragma unroll
    for (int i = 0; i < 4; ++i) {
        // A -> k-pair interleaved: two b64 stores, lanes at stride-2 dwords
        // (consecutive am) cover all 64 banks -> conflict-free.
        const int rp = (akb + 2 * i) * 2;
        v2f p0 = {ra[i][0], ra[i][1]};
        v2f p1 = {ra[i][2], ra[i][3]};
        *(v2f*)(Al + (size_t)(rp + 0) * PITCH + 2 * am) = p0;
        *(v2f*)(Al + (size_t)(rp + 1) * PITCH + 2 * am) = p1;
    }
#pragma unroll
    for (int i = 0; i < 2; ++i) {
        // B -> k-pair interleaved: pack {lo,hi} pairs, 32B contiguous per thread.
        const int rp = brp0 + 8 * i;
        f4 w0 = {rbl[i][0], rbh[i][0], rbl[i][1], rbh[i][1]};
        f4 w1 = {rbl[i][2], rbh[i][2], rbl[i][3], rbh[i][3]};
        float* dst = Bl + (size_t)rp * PITCH + bnq * 8;
        *(f4*)(dst) = w0;
        *(f4*)(dst + 4) = w1;
    }
}

__global__ __launch_bounds__(256) void gg_wmma_f32_kernel(GGArgs p, float* __restrict__ out) {
    __shared__ float Al[2][(BK / 2) * PITCH];   // 2 x 18432 B
    __shared__ float Bl[2][(BK / 2) * PITCH];   // 2 x 18432 B

    const int bid = blockIdx.x;
    // Uniform scalar lookup: which problem does this tile belong to?
    int g = 0;
#pragma unroll
    for (int t = 0; t < 7; ++t)
        if (bid >= p.tileBase[t + 1]) g = t + 1;

    const int local = bid - p.tileBase[g];
    const int tn = local & 31;   // N/BN == 32 tiles along N for every problem
    const int tm = local >> 5;
    const float* __restrict__ A = p.A[g];
    const float* __restrict__ B = p.B[g];
    float* __restrict__ C = out + p.outOff[g];
    const int m0 = tm * BM;
    const int n0 = tn * BN;

    const int tid    = threadIdx.x;
    const int lane   = tid & 31;
    const int wid    = tid >> 5;       // 8 waves
    const int lane15 = lane & 15;
    const int laneHi = lane >> 4;      // 0: K={0,1}, 1: K={2,3} (ISA 16x4 layouts)
    const int wm     = wid >> 2;       // 0..1  -> 64-row slab
    const int wn     = wid & 3;        // 0..3  -> 32-col slab
    const int mwb    = wm * 64;
    const int nwb    = wn * 32;

    // Global-load index decomposition
    const int am   = tid & 127;  // A row within tile
    const int akb  = tid >> 7;   // 0/1 -> which float4 column group
    const int bnq  = tid & 31;   // B float4 group along n
    const int brp0 = tid >> 5;   // 0..7 -> B k-pair row base

    v8f acc[4][2];
#pragma unroll
    for (int i = 0; i < 4; ++i)
#pragma unroll
        for (int j = 0; j < 2; ++j)
            acc[i][j] = (v8f){0.f, 0.f, 0.f, 0.f, 0.f, 0.f, 0.f, 0.f};

    f4 ra[4], rbl[2], rbh[2];

    // Prologue: stage k-block 0
    load_tiles(A, B, m0, n0, 0, am, akb, bnq, brp0, ra, rbl, rbh);
    store_tiles(&Al[0][0], &Bl[0][0], am, akb, bnq, brp0, ra, rbl, rbh);
    __syncthreads();

    for (int kb = 0; kb < NKB; ++kb) {
        const int buf = kb & 1;
        const float* __restrict__ Ab = &Al[buf][0];
        const float* __restrict__ Bb = &Bl[buf][0];

        // Prefetch next k-block into registers while we compute this one
        if (kb + 1 < NKB)
            load_tiles(A, B, m0, n0, (kb + 1) * BK, am, akb, bnq, brp0, ra, rbl, rbh);

        // 8 K-steps of 4, 8 WMMAs each (4 m-tiles x 2 n-tiles).
        // Fragments are double-buffered: loads for step kk+1 are issued before
        // the WMMAs of step kk so the dscnt wait lands behind a full WMMA group.
        v2f af[2][4], bf[2][2];
        {
            const int r0 = laneHi;
#pragma unroll
            for (int i = 0; i < 4; ++i)
                af[0][i] = *(const v2f*)(Ab + (size_t)r0 * PITCH +
                                         2 * (mwb + i * 16 + lane15));
#pragma unroll
            for (int j = 0; j < 2; ++j)
                bf[0][j] = *(const v2f*)(Bb + (size_t)r0 * PITCH +
                                         2 * (nwb + j * 16 + lane15));
        }
#pragma unroll
        for (int kk = 0; kk < BK / 4; ++kk) {
            const int cur = kk & 1;
            const int nxt = cur ^ 1;
            if (kk + 1 < BK / 4) {
                const int r = 2 * (kk + 1) + laneHi;   // next k-pair row
#pragma unroll
                for (int i = 0; i < 4; ++i)
                    af[nxt][i] = *(const v2f*)(Ab + (size_t)r * PITCH +
                                               2 * (mwb + i * 16 + lane15));
#pragma unroll
                for (int j = 0; j < 2; ++j)
                    bf[nxt][j] = *(const v2f*)(Bb + (size_t)r * PITCH +
                                               2 * (nwb + j * 16 + lane15));
            }
#pragma unroll
            for (int i = 0; i < 4; ++i)
#pragma unroll
                for (int j = 0; j < 2; ++j)
                    acc[i][j] = __builtin_amdgcn_wmma_f32_16x16x4_f32(
                        /*neg_a=*/false, af[cur][i],
                        /*neg_b=*/false, bf[cur][j],
                        /*c_mod=*/(short)0, acc[i][j],
                        /*reuse_a=*/false, /*reuse_b=*/false);
        }

        // Ping-pong: write next buffer, one barrier per iteration is sufficient
        if (kb + 1 < NKB) {
            store_tiles(&Al[buf ^ 1][0], &Bl[buf ^ 1][0], am, akb, bnq, brp0,
                        ra, rbl, rbh);
            __syncthreads();
        }
    }

    // Epilogue: C/D layout — VGPR v holds M=v (lanes 0-15) and M=v+8 (lanes 16-31)
#pragma unroll
    for (int i = 0; i < 4; ++i)
#pragma unroll
        for (int j = 0; j < 2; ++j) {
            const int row0 = m0 + mwb + i * 16 + laneHi * 8;
            const int col  = n0 + nwb + j * 16 + lane15;
#pragma unroll
            for (int v = 0; v < 8; ++v)
                C[(size_t)(row0 + v) * GG_N + col] = acc[i][j][v];
        }
}

extern "C" void kernel_launch(void* const* d_in, const int* in_sizes, int n_in,
                              void* d_out, int out_size, void* d_ws, size_t ws_size,
                              hipStream_t stream) {
    (void)n_in; (void)out_size; (void)d_ws; (void)ws_size;
    GGArgs p;
    int tiles = 0;
    unsigned long long off = 0;
    for (int i = 0; i < 8; ++i) {
        p.A[i] = (const float*)d_in[2 * i];       // setup_inputs order: a0,b0,a1,b1,...
        p.B[i] = (const float*)d_in[2 * i + 1];
        const int M = in_sizes[2 * i] / GG_K;     // all M are multiples of 128
        p.tileBase[i] = tiles;
        tiles += (M / BM) * (GG_N / BN);
        p.outOff[i] = off;
        off += (unsigned long long)M * (unsigned long long)GG_N;
    }
    p.tileBase[8] = tiles;

    hipLaunchKernelGGL(gg_wmma_f32_kernel, dim3(tiles), dim3(256), 0, stream,
                       p, (float*)d_out);
}